// NECTAR_Binning_45286135169515
// MI455X (gfx1250) — compile-verified
//
#include <hip/hip_runtime.h>
#include <hip/hip_bf16.h>
#include <cstdint>

// Problem constants (fixed by the reference)
#define NUM_CLASSES 4
#define IMG_H 512
#define IMG_W 512
#define NUM_BINS 15
#define NBR 9                      // K*K neighbor-count states, counts in [0,8]

// Tiling: 64x32 interior, 1-pixel halo. Halo row = 66 pixels mapped to LDS
// columns 3..68 so the 64-wide interior (cols 4..67) is 16B aligned.
#define TILE_H 32
#define TILE_W 64
#define HALO_H (TILE_H + 2)        // 34
#define HALO_W (TILE_W + 2)        // 66
#define COL0 3                     // LDS column of halo-left (w0-1)
#define LDS_PITCH 72               // floats per LDS row (72*4 = 288B, 16B multiple)
#define NTHREADS 256

// gfx1250 async global->LDS path (ASYNCcnt-tracked). Fallbacks compile anywhere.
#if defined(__AMDGCN__) && __has_builtin(__builtin_amdgcn_global_load_async_to_lds_b32)
#define USE_ASYNC_LDS 1
#else
#define USE_ASYNC_LDS 0
#endif
#if defined(__AMDGCN__) && __has_builtin(__builtin_amdgcn_global_load_async_to_lds_b128)
#define USE_ASYNC_B128 1
#else
#define USE_ASYNC_B128 0
#endif

#if USE_ASYNC_LDS
typedef int v4i_t __attribute__((ext_vector_type(4)));
// AS1: generic pointers to global memory share the same bit pattern.
// AS3: LLVM lowers generic->local addrspacecast as truncation to low 32 bits.
#define TO_GLOBAL(p)  ((__attribute__((address_space(1))) int*)(uintptr_t)(p))
#define TO_LDS(p)     ((__attribute__((address_space(3))) int*)(uint32_t)(uintptr_t)(p))
#define TO_GLOBAL4(p) ((__attribute__((address_space(1))) v4i_t*)(uintptr_t)(p))
#define TO_LDS4(p)    ((__attribute__((address_space(3))) v4i_t*)(uint32_t)(uintptr_t)(p))
#endif

__global__ __launch_bounds__(NTHREADS)
void NECTAR_Binning_45286135169515_kernel(const float* __restrict__ logits,
                                          const float* __restrict__ val_freqs,
                                          float* __restrict__ out)
{
    __shared__ __align__(16) float s_lg[NUM_CLASSES][HALO_H][LDS_PITCH]; // halo'd logits
    __shared__ unsigned char       s_cls[HALO_H][LDS_PITCH];             // argmax class
    __shared__ float               s_vf[NUM_CLASSES * NBR * NUM_BINS];   // calib table

    const int tid = threadIdx.x;
    const int b   = blockIdx.z;
    const int h0  = blockIdx.y * TILE_H;
    const int w0  = blockIdx.x * TILE_W;

    // calibration table -> LDS
    for (int i = tid; i < NUM_CLASSES * NBR * NUM_BINS; i += NTHREADS)
        s_vf[i] = val_freqs[i];

    const int rows = NUM_CLASSES * HALO_H;             // 136 (class,haloRow) rows

    // ---- Stage A1: aligned 64-wide interior of each halo row, B128 asyncs ----
    // 16 lanes x 16B per row; both global and LDS addresses 16B aligned.
    for (int p = tid; p < rows * 16; p += NTHREADS) {
        int row = p >> 4;                              // 0..135
        int l4  = p & 15;
        int c   = row / HALO_H;
        int r   = row - c * HALO_H;
        int h   = h0 - 1 + r;
        if ((unsigned)h < IMG_H) {
            int w   = w0 + (l4 << 2);                  // always in [0,511]
            int off = (((b * NUM_CLASSES + c) * IMG_H) + h) * IMG_W + w;
            float*       dst = &s_lg[c][r][COL0 + 1 + (l4 << 2)];
            const float* src = logits + off;
#if USE_ASYNC_B128
            __builtin_amdgcn_global_load_async_to_lds_b128(TO_GLOBAL4(src), TO_LDS4(dst), 0, 0);
#elif USE_ASYNC_LDS
            __builtin_amdgcn_global_load_async_to_lds_b32(TO_GLOBAL(src + 0), TO_LDS(dst + 0), 0, 0);
            __builtin_amdgcn_global_load_async_to_lds_b32(TO_GLOBAL(src + 1), TO_LDS(dst + 1), 0, 0);
            __builtin_amdgcn_global_load_async_to_lds_b32(TO_GLOBAL(src + 2), TO_LDS(dst + 2), 0, 0);
            __builtin_amdgcn_global_load_async_to_lds_b32(TO_GLOBAL(src + 3), TO_LDS(dst + 3), 0, 0);
#else
            dst[0] = src[0]; dst[1] = src[1]; dst[2] = src[2]; dst[3] = src[3];
#endif
        }
    }

    // ---- Stage A2: left/right halo edge columns, B32 asyncs ----
    for (int p = tid; p < rows * 2; p += NTHREADS) {
        int row  = p >> 1;
        int side = p & 1;
        int c    = row / HALO_H;
        int r    = row - c * HALO_H;
        int h    = h0 - 1 + r;
        int w    = side ? (w0 + TILE_W) : (w0 - 1);
        int col  = side ? (COL0 + 1 + TILE_W) : COL0;
        if ((unsigned)h < IMG_H && (unsigned)w < IMG_W) {
            int off = (((b * NUM_CLASSES + c) * IMG_H) + h) * IMG_W + w;
#if USE_ASYNC_LDS
            __builtin_amdgcn_global_load_async_to_lds_b32(
                TO_GLOBAL(logits + off), TO_LDS(&s_lg[c][r][col]), 0, 0);
#else
            s_lg[c][r][col] = logits[off];
#endif
        }
        // OOB halo cells left uninitialized: stage B marks them 0xFF and
        // stage C never reads their logits.
    }

#if USE_ASYNC_LDS
#if __has_builtin(__builtin_amdgcn_s_wait_asynccnt)
    __builtin_amdgcn_s_wait_asynccnt(0);               // wave's async copies done
#else
    asm volatile("s_wait_asynccnt 0" ::: "memory");
#endif
#endif
    __syncthreads();                                   // all waves' copies visible

    // ---- Stage B: argmax class per halo pixel (first-max tie rule) ----
    const int plane = HALO_H * HALO_W;                 // 2244
    for (int p = tid; p < plane; p += NTHREADS) {
        int r = p / HALO_W;
        int k = p - r * HALO_W;                        // 0..65
        int h = h0 - 1 + r;
        int w = w0 - 1 + k;
        unsigned char cls = 0xFF;                      // OOB => matches no class (zero pad)
        if ((unsigned)h < IMG_H && (unsigned)w < IMG_W) {
            float best = s_lg[0][r][COL0 + k];
            int   bi   = 0;
#pragma unroll
            for (int c = 1; c < NUM_CLASSES; ++c) {
                float v = s_lg[c][r][COL0 + k];
                if (v > best) { best = v; bi = c; }    // strict > keeps first max
            }
            cls = (unsigned char)bi;
        }
        s_cls[r][COL0 + k] = cls;
    }
    __syncthreads();

    // ---- Stage C: stencil counts + softmax + bin + gather + normalize ----
    for (int q = tid; q < TILE_H * TILE_W; q += NTHREADS) {
        int r  = q >> 6;                               // / TILE_W
        int cc = q & (TILE_W - 1);
        int lr = r + 1;                                // halo row
        int lc = cc + COL0 + 1;                        // halo col

        // branchless same-class neighbor counts (8 neighbors)
        int cnt0 = 0, cnt1 = 0, cnt2 = 0, cnt3 = 0;
#pragma unroll
        for (int dr = -1; dr <= 1; ++dr) {
#pragma unroll
            for (int dc = -1; dc <= 1; ++dc) {
                if (dr == 0 && dc == 0) continue;
                unsigned v = s_cls[lr + dr][lc + dc];
                cnt0 += (v == 0u); cnt1 += (v == 1u);
                cnt2 += (v == 2u); cnt3 += (v == 3u);
            }
        }
        const int cnt[NUM_CLASSES] = {cnt0, cnt1, cnt2, cnt3};

        float l[NUM_CLASSES];
#pragma unroll
        for (int c = 0; c < NUM_CLASSES; ++c) l[c] = s_lg[c][lr][lc];

        float m = fmaxf(fmaxf(l[0], l[1]), fmaxf(l[2], l[3]));
        float e[NUM_CLASSES];
        float es = 0.0f;
#pragma unroll
        for (int c = 0; c < NUM_CLASSES; ++c) { e[c] = __expf(l[c] - m); es += e[c]; }
        float inv = 1.0f / es;

        float cal[NUM_CLASSES];
        float csum = 0.0f;
#pragma unroll
        for (int c = 0; c < NUM_CLASSES; ++c) {
            float pc  = e[c] * inv;                    // softmax prob in [0,1]
            int   bin = (int)(pc * (float)NUM_BINS);   // trunc == floor for pc>=0
            bin = bin > (NUM_BINS - 1) ? (NUM_BINS - 1) : (bin < 0 ? 0 : bin);
            cal[c] = s_vf[(c * NBR + cnt[c]) * NUM_BINS + bin];
            csum  += cal[c];
        }
        if (csum == 0.0f) csum = 1.0f;                 // zero-sum guard
        float icn = 1.0f / csum;

        int h = h0 + r, w = w0 + cc;
#pragma unroll
        for (int c = 0; c < NUM_CLASSES; ++c) {
            int off = (((b * NUM_CLASSES + c) * IMG_H) + h) * IMG_W + w;
            out[off] = cal[c] * icn;                   // coalesced per class plane
        }
    }
}

extern "C" void kernel_launch(void* const* d_in, const int* in_sizes, int n_in,
                              void* d_out, int out_size, void* d_ws, size_t ws_size,
                              hipStream_t stream) {
    const float* logits    = (const float*)d_in[0];   // [B,4,512,512] f32
    const float* val_freqs = (const float*)d_in[1];   // [4,9,15] f32
    float* out             = (float*)d_out;           // [B,4,512,512] f32
    (void)d_ws; (void)ws_size; (void)n_in; (void)out_size;

    const int B = in_sizes[0] / (NUM_CLASSES * IMG_H * IMG_W);
    dim3 grid(IMG_W / TILE_W, IMG_H / TILE_H, B);     // (8, 16, 16)
    dim3 block(NTHREADS);
    NECTAR_Binning_45286135169515_kernel<<<grid, block, 0, stream>>>(logits, val_freqs, out);
}